// PanopticHead_32710470926809
// MI455X (gfx1250) — compile-verified
//
#include <hip/hip_runtime.h>
#include <stdint.h>
#include <math.h>

#define NB 4
#define KK 32
#define HH 384
#define WW 384
#define MM 28
#define STUFFC 53
#define THINGC 80
#define C_IN  (STUFFC + THINGC)   // 133
#define C_OUT (STUFFC + KK)       // 85
#define HW (HH * WW)              // 147456
#define IGNORE_LBL 255
#define PX_PER_THREAD 4
#define BLOCK 256
#define PX_PER_BLOCK (BLOCK * PX_PER_THREAD)   // 1024
#define BLOCKS_PER_IMG (HW / PX_PER_BLOCK)     // 144
#define MASK_ELEMS (KK * MM * MM)              // 25088 f32 = 100352 B

typedef __attribute__((ext_vector_type(4))) unsigned int v4u;
typedef __attribute__((ext_vector_type(8))) int          v8i;
typedef __attribute__((ext_vector_type(4))) int          v4i;

__global__ __launch_bounds__(BLOCK)
void pan_main(const float* __restrict__ sem, const float* __restrict__ maskp,
              const float* __restrict__ boxes, const int* __restrict__ clsi,
              const int* __restrict__ gtp, float* __restrict__ out,
              float* __restrict__ partial)
{
    __shared__ float s_mask[MASK_ELEMS];   // all K masks for this image
    __shared__ float s_box[KK * 4];
    __shared__ int   s_cls[KK];
    __shared__ float s_red[(BLOCK / 32) * 2];

    const int n   = blockIdx.y;
    const int tid = threadIdx.x;

    // ---- TDM: async bulk copy of the 100KB mask set into LDS (wave 0 issues) ----
    if (tid < 32) {
        const uint64_t gaddr = (uint64_t)(uintptr_t)(maskp + (size_t)n * MASK_ELEMS);
        const uint32_t laddr = (uint32_t)(uintptr_t)(&s_mask[0]); // LDS aperture: low 32b = LDS byte addr
        // D# group 0: count=1 | lds_addr | global_addr[56:0] | type=2
        v4u g0 = { 1u,
                   laddr,
                   (uint32_t)(gaddr & 0xFFFFFFFFu),
                   (uint32_t)((gaddr >> 32) & 0x01FFFFFFu) | 0x80000000u };
        // D# group 1: data_size=4B; tensor_dim0=25088; tensor_dim1=1;
        //             tile_dim0=25088; tile_dim1=1; tensor_dim0_stride=25088
        v8i g1 = { (int)(2u << 16),
                   (int)((uint32_t)(MASK_ELEMS & 0xFFFF) << 16),
                   (int)(((uint32_t)MASK_ELEMS >> 16) | (1u << 16)),
                   (int)((uint32_t)MASK_ELEMS << 16),
                   1,
                   (int)MASK_ELEMS,
                   0, 0 };
        v4i gz4 = { 0, 0, 0, 0 };
        v8i gz8 = { 0, 0, 0, 0, 0, 0, 0, 0 };
        // clang-23 / therock-10.0 form: 6 args (extra int32x8 operand, zero-filled)
        __builtin_amdgcn_tensor_load_to_lds(g0, g1, gz4, gz4, gz8, 0);
    }
    // Stage boxes / class ids while the TDM runs
    if (tid < KK * 4) s_box[tid] = boxes[n * KK * 4 + tid];
    if (tid < KK)     s_cls[tid] = clsi[n * KK + tid];
    __builtin_amdgcn_s_wait_tensorcnt(0);   // no-op for non-issuing waves
    __syncthreads();

    // ---- Per-thread: 4 consecutive pixels of one row ----
    const int    pbase  = blockIdx.x * PX_PER_BLOCK + tid * PX_PER_THREAD;
    const int    y      = pbase / WW;
    const int    x0     = pbase % WW;
    const size_t in_px  = (size_t)n * C_IN  * HW + pbase;
    const size_t out_px = (size_t)n * C_OUT * HW + pbase;

    const int4 gtv = *(const int4*)(gtp + (size_t)n * HW + pbase);
    int gtl[4] = { gtv.x, gtv.y, gtv.z, gtv.w };

    float m[4], ssum[4], vg[4];
    // channel 0 initializes the streaming log-sum-exp
    {
        const float4 v = *(const float4*)(sem + in_px);
        *(float4*)(out + out_px) = v;
        const float vv[4] = { v.x, v.y, v.z, v.w };
#pragma unroll
        for (int j = 0; j < 4; ++j) {
            m[j] = vv[j]; ssum[j] = 1.0f;
            vg[j] = (gtl[j] == 0) ? vv[j] : 0.0f;
        }
    }

    auto upd = [&](int c, float4 v) {
        const float vv[4] = { v.x, v.y, v.z, v.w };
#pragma unroll
        for (int j = 0; j < 4; ++j) {
            const float mn = fmaxf(m[j], vv[j]);
            ssum[j] = ssum[j] * __expf(m[j] - mn) + __expf(vv[j] - mn);
            m[j] = mn;
            if (gtl[j] == c) vg[j] = vv[j];
        }
    };

    // ---- Stuff channels: straight copy + LSE update (b128 in, b128 out) ----
    for (int c = 1; c < STUFFC; ++c) {
        const float4 v = *(const float4*)(sem + in_px + (size_t)c * HW);
        *(float4*)(out + out_px + (size_t)c * HW) = v;
        upd(c, v);
    }

    // ---- Thing channels: bilinear paste from LDS + cropped semantic channel ----
    const float yc = (float)y + 0.5f;
    for (int k = 0; k < KK; ++k) {
        const float bx1 = s_box[k * 4 + 0], by1 = s_box[k * 4 + 1];
        const float bx2 = s_box[k * 4 + 2], by2 = s_box[k * 4 + 3];
        float4 v = make_float4(0.f, 0.f, 0.f, 0.f);
        if (yc >= by1 && yc <= by2) {           // wave-uniform (one row per wave)
            const float myv = (yc - by1) / (by2 - by1) * (float)MM - 0.5f;
            const float yf  = floorf(myv);
            const float wy  = myv - yf;
            int iy0 = (int)yf; iy0 = iy0 < 0 ? 0 : (iy0 > MM - 1 ? MM - 1 : iy0);
            const int iy1 = (iy0 + 1 > MM - 1) ? MM - 1 : iy0 + 1;
            const float* mk0 = s_mask + k * MM * MM + iy0 * MM;
            const float* mk1 = s_mask + k * MM * MM + iy1 * MM;
            const float4 crop = *(const float4*)(sem + in_px + (size_t)(STUFFC + s_cls[k]) * HW);
            const float cr[4] = { crop.x, crop.y, crop.z, crop.w };
            const float invw = 1.0f / (bx2 - bx1);
            float vv[4];
#pragma unroll
            for (int j = 0; j < 4; ++j) {
                const float xc = (float)(x0 + j) + 0.5f;
                float r = 0.0f;
                if (xc >= bx1 && xc <= bx2) {
                    const float mxv = (xc - bx1) * invw * (float)MM - 0.5f;
                    const float xf  = floorf(mxv);
                    const float wx  = mxv - xf;
                    int ix0 = (int)xf; ix0 = ix0 < 0 ? 0 : (ix0 > MM - 1 ? MM - 1 : ix0);
                    const int ix1 = (ix0 + 1 > MM - 1) ? MM - 1 : ix0 + 1;
                    const float row0 = mk0[ix0] * (1.0f - wy) + mk1[ix0] * wy;
                    const float row1 = mk0[ix1] * (1.0f - wy) + mk1[ix1] * wy;
                    r = row0 * (1.0f - wx) + row1 * wx + cr[j];
                }
                vv[j] = r;
            }
            v = make_float4(vv[0], vv[1], vv[2], vv[3]);
        }
        *(float4*)(out + out_px + (size_t)(STUFFC + k) * HW) = v;
        upd(STUFFC + k, v);
    }

    // ---- Loss: nll = logsumexp - v_gt ; deterministic block reduction ----
    float lsum = 0.0f, lcnt = 0.0f;
#pragma unroll
    for (int j = 0; j < 4; ++j) {
        if (gtl[j] != IGNORE_LBL) {
            lsum += (m[j] + __logf(ssum[j])) - vg[j];
            lcnt += 1.0f;
        }
    }
    for (int off = 16; off > 0; off >>= 1) {       // wave32 reduce
        lsum += __shfl_down(lsum, off, 32);
        lcnt += __shfl_down(lcnt, off, 32);
    }
    const int wid = tid >> 5;
    if ((tid & 31) == 0) { s_red[wid * 2] = lsum; s_red[wid * 2 + 1] = lcnt; }
    __syncthreads();
    if (tid == 0) {
        float ts = 0.0f, tc = 0.0f;
        for (int w = 0; w < BLOCK / 32; ++w) { ts += s_red[w * 2]; tc += s_red[w * 2 + 1]; }
        const int bid = blockIdx.y * gridDim.x + blockIdx.x;
        partial[bid * 2] = ts; partial[bid * 2 + 1] = tc;
    }
}

__global__ __launch_bounds__(256)
void pan_finalize(const float* __restrict__ partial, float* __restrict__ out_loss, int nblocks)
{
    __shared__ float rs[256], rc[256];
    float ts = 0.0f, tc = 0.0f;
    for (int i = threadIdx.x; i < nblocks; i += 256) {
        ts += partial[i * 2]; tc += partial[i * 2 + 1];
    }
    rs[threadIdx.x] = ts; rc[threadIdx.x] = tc;
    __syncthreads();
    for (int off = 128; off > 0; off >>= 1) {
        if ((int)threadIdx.x < off) {
            rs[threadIdx.x] += rs[threadIdx.x + off];
            rc[threadIdx.x] += rc[threadIdx.x + off];
        }
        __syncthreads();
    }
    if (threadIdx.x == 0) out_loss[0] = rs[0] / fmaxf(rc[0], 1.0f);
}

extern "C" void kernel_launch(void* const* d_in, const int* in_sizes, int n_in,
                              void* d_out, int out_size, void* d_ws, size_t ws_size,
                              hipStream_t stream) {
    const float* sem   = (const float*)d_in[0];  // (4,133,384,384) f32
    const float* maskp = (const float*)d_in[1];  // (4,32,28,28)    f32
    const float* boxes = (const float*)d_in[2];  // (4,32,4)        f32
    const int*   clsi  = (const int*)  d_in[3];  // (4,32)          i32
    const int*   gtp   = (const int*)  d_in[4];  // (4,384,384)     i32
    float* out     = (float*)d_out;              // 4*85*384*384 logits + 1 loss
    float* partial = (float*)d_ws;               // 576 * {sum,cnt} = 4608 B

    dim3 grid(BLOCKS_PER_IMG, NB);
    pan_main<<<grid, BLOCK, 0, stream>>>(sem, maskp, boxes, clsi, gtp, out, partial);

    const int nblocks = BLOCKS_PER_IMG * NB;
    pan_finalize<<<1, 256, 0, stream>>>(partial, out + (size_t)NB * C_OUT * HW, nblocks);
}